// DotProductAttention_10462540333674
// MI455X (gfx1250) — compile-verified
//
#include <hip/hip_runtime.h>

// ---------------------------------------------------------------------------
// Flash attention (fp32, CDNA5 WMMA + TDM) for B=8, Q=2048, K=2048, D=128
// with per-batch valid-length key masking (MASK_FILL = -1e6).
//
// Round 3: Tensor Data Mover (tensor_load_to_lds) stages K/V tiles into
// double-buffered LDS (one DMA instruction per tile per operand, issued by
// wave 0, synced with s_wait_tensorcnt + one barrier per iteration). This
// removes the 32 staging VGPRs (back under the 256-VGPR direct window) and
// the explicit LDS store pass. Compute: 32-wide key tiles, 4 independent
// QK WMMA chains, exp2-domain online softmax, per-wave P transpose.
// ---------------------------------------------------------------------------

typedef __attribute__((ext_vector_type(2))) float        v2f;
typedef __attribute__((ext_vector_type(8))) float        v8f;
typedef __attribute__((ext_vector_type(4))) unsigned int v4u;
typedef __attribute__((ext_vector_type(8))) int          v8i;
typedef __attribute__((ext_vector_type(4))) int          v4i;

constexpr int Bn = 8;
constexpr int Qn = 2048;
constexpr int Kn = 2048;
constexpr int Dn = 128;

constexpr int WAVES  = 8;            // waves per block
constexpr int QTILE  = WAVES * 16;   // 128 query rows per block
constexpr int KTILE  = 32;           // keys per iteration (2 x WMMA N)
constexpr int DSTEPS = Dn / 4;       // 32 K-steps of 16x16x4 per score tile
constexpr int PSTEPS = KTILE / 4;    // 8  K-steps for the PV GEMM

constexpr float LOG2E    = 1.44269504088896340736f;
constexpr float SCALE_L2 = 0.08838834764831845f * LOG2E;  // (1/sqrt(128))*log2e
constexpr float MASK_L2  = -1.0e6f;  // huge negative in exp2 domain -> 0
constexpr float NEG_BIG  = -3.0e38f;

// ---- Tensor DMA: move one rows x 128 fp32 tile (row-major) global -> LDS. --
// D# per CDNA5 ISA ch.8: group0 = {count, lds_addr, global_addr, type=2},
// group1 = {wg_mask=0, data_size=4B, tensor dims, tile dims, dim0 stride}.
__device__ __forceinline__ void tdm_load_tile(const float* gsrc,
                                              unsigned lds_off, int rows) {
    const unsigned long long ga = (unsigned long long)(uintptr_t)gsrc;
    v4u g0;
    g0[0] = 1u;                                          // count=1, user mode
    g0[1] = lds_off;                                     // LDS byte address
    g0[2] = (unsigned)(ga & 0xFFFFFFFFu);                // global_addr[31:0]
    g0[3] = (unsigned)((ga >> 32) & 0x01FFFFFFu)         // global_addr[56:32]
          | (2u << 30);                                  // type=2 ("image")
    v8i g1;
    g1[0] = 0x00020000;          // workgroup_mask=0, data_size=2 (4 bytes)
    g1[1] = (int)((unsigned)Dn << 16);    // tensor_dim0 = 128 (bits 79:48 lo)
    g1[2] = (int)((unsigned)rows << 16);  // dim0 hi = 0 ; tensor_dim1 = rows
    g1[3] = (int)((unsigned)Dn << 16);    // dim1 hi = 0 ; tile_dim0 = 128
    g1[4] = rows;                         // tile_dim1 = rows ; tile_dim2 = 0
    g1[5] = Dn;                           // tensor_dim0_stride = 128
    g1[6] = 0;
    g1[7] = 0;
    const v4i z4 = (v4i)0;
#if __clang_major__ >= 23
    const v8i z8 = (v8i)0;
    __builtin_amdgcn_tensor_load_to_lds(g0, g1, z4, z4, z8, 0);
#else
    __builtin_amdgcn_tensor_load_to_lds(g0, g1, z4, z4, 0);
#endif
}

__global__ __launch_bounds__(256, 1)
void fattn_f32_wmma(const float* __restrict__ Qp,
                    const float* __restrict__ Kp,
                    const float* __restrict__ Vp,
                    const int*   __restrict__ VLp,
                    float*       __restrict__ Op)
{
    // Double-buffered K and V tiles (row-major [r][d]) + per-wave P scratch.
    __shared__ __align__(16) float sK[2][KTILE * Dn];     // 2 x 16 KB
    __shared__ __align__(16) float sV[2][KTILE * Dn];     // 2 x 16 KB
    __shared__ __align__(16) float sP[WAVES][16 * KTILE]; // 16 KB

    const int tid   = threadIdx.x;
    const int wave  = tid >> 5;
    const int lane  = tid & 31;
    const int lhalf = lane >> 4;   // 0: lanes 0-15, 1: lanes 16-31
    const int lmod  = lane & 15;

    const int b    = blockIdx.x / (Qn / QTILE);
    const int qblk = blockIdx.x % (Qn / QTILE);
    const int q0   = qblk * QTILE + wave * 16;   // this wave's 16 query rows

    const int valid  = VLp[b];
    const int ntiles = min(Kn / KTILE, (valid + KTILE - 1) / KTILE);

    const float* kbatch = Kp + (size_t)b * Kn * Dn;
    const float* vbatch = Vp + (size_t)b * Kn * Dn;

    // ---- Load Q A-fragments (16x128 stripe), pre-scaled into exp2 domain. --
    v2f qa[DSTEPS];
    {
        const float* qbase = Qp + ((size_t)b * Qn + q0 + lmod) * Dn + 2 * lhalf;
#pragma unroll
        for (int s = 0; s < DSTEPS; ++s) {
            v2f t = *(const v2f*)(qbase + 4 * s);
            qa[s] = t * SCALE_L2;
        }
    }

    // ---- Online-softmax state + output accumulators (16 x 128 stripe). ----
    v8f acc[8];
    float mrow[8], lrow[8];
#pragma unroll
    for (int c = 0; c < 8; ++c) acc[c] = (v8f)0.0f;
#pragma unroll
    for (int i = 0; i < 8; ++i) { mrow[i] = NEG_BIG; lrow[i] = 0.0f; }

    float* pp = &sP[wave][0];

    // ---- Pipeline prologue: TDM tile 0 into buffer 0. ----
    if (wave == 0 && ntiles > 0) {
        tdm_load_tile(kbatch, (unsigned)(uintptr_t)&sK[0][0], KTILE);
        tdm_load_tile(vbatch, (unsigned)(uintptr_t)&sV[0][0], KTILE);
        __builtin_amdgcn_s_wait_tensorcnt(0);
    }
    __syncthreads();

    for (int kt = 0; kt < ntiles; ++kt) {
        const int  cur       = kt & 1;
        const bool have_next = (kt + 1) < ntiles;

        // Issue next tile's DMA into the other buffer; safe: its last readers
        // finished before the previous iteration's barrier.
        if (wave == 0 && have_next) {
            const size_t goff = (size_t)(kt + 1) * KTILE * Dn;
            tdm_load_tile(kbatch + goff,
                          (unsigned)(uintptr_t)&sK[1 - cur][0], KTILE);
            tdm_load_tile(vbatch + goff,
                          (unsigned)(uintptr_t)&sV[1 - cur][0], KTILE);
        }

        // ---- S = (Q*scale) . K^T : two 16-wide score tiles, 4 indep chains.
        const float* kb = &sK[cur][lmod * Dn + 2 * lhalf];
        v8f c0a = (v8f)0.0f, c0b = (v8f)0.0f;
        v8f c1a = (v8f)0.0f, c1b = (v8f)0.0f;
#pragma unroll
        for (int s = 0; s < DSTEPS; s += 2) {
            v2f b0 = *(const v2f*)(kb + 4 * s);
            v2f b1 = *(const v2f*)(kb + 4 * (s + 1));
            c0a = __builtin_amdgcn_wmma_f32_16x16x4_f32(
                      false, qa[s],     false, b0, (short)0, c0a, false, false);
            c0b = __builtin_amdgcn_wmma_f32_16x16x4_f32(
                      false, qa[s + 1], false, b1, (short)0, c0b, false, false);
        }
        const float* kb1 = kb + 16 * Dn;
#pragma unroll
        for (int s = 0; s < DSTEPS; s += 2) {
            v2f b0 = *(const v2f*)(kb1 + 4 * s);
            v2f b1 = *(const v2f*)(kb1 + 4 * (s + 1));
            c1a = __builtin_amdgcn_wmma_f32_16x16x4_f32(
                      false, qa[s],     false, b0, (short)0, c1a, false, false);
            c1b = __builtin_amdgcn_wmma_f32_16x16x4_f32(
                      false, qa[s + 1], false, b1, (short)0, c1b, false, false);
        }

        // ---- Length mask (per key column), exp2-domain scores. ----
        const bool m0 = (kt * KTILE + lmod)      >= valid;
        const bool m1 = (kt * KTILE + 16 + lmod) >= valid;
        float sv0[8], sv1[8];
#pragma unroll
        for (int i = 0; i < 8; ++i) {
            sv0[i] = m0 ? MASK_L2 : (c0a[i] + c0b[i]);
            sv1[i] = m1 ? MASK_L2 : (c1a[i] + c1b[i]);
        }

        // ---- Row max across the 16-lane group holding each row. ----
        float rmax[8];
#pragma unroll
        for (int i = 0; i < 8; ++i) rmax[i] = fmaxf(sv0[i], sv1[i]);
#pragma unroll
        for (int off = 1; off < 16; off <<= 1) {
#pragma unroll
            for (int i = 0; i < 8; ++i)
                rmax[i] = fmaxf(rmax[i], __shfl_xor(rmax[i], off, 32));
        }

        // ---- Online softmax update (exp2 domain). ----
        float fac[8], p0[8], p1[8];
#pragma unroll
        for (int i = 0; i < 8; ++i) {
            const float mnew = fmaxf(mrow[i], rmax[i]);
            fac[i]  = __builtin_amdgcn_exp2f(mrow[i] - mnew);
            mrow[i] = mnew;
            p0[i]   = __builtin_amdgcn_exp2f(sv0[i] - mnew);
            p1[i]   = __builtin_amdgcn_exp2f(sv1[i] - mnew);
        }
        float rsum[8];
#pragma unroll
        for (int i = 0; i < 8; ++i) rsum[i] = p0[i] + p1[i];
#pragma unroll
        for (int off = 1; off < 16; off <<= 1) {
#pragma unroll
            for (int i = 0; i < 8; ++i)
                rsum[i] += __shfl_xor(rsum[i], off, 32);
        }
#pragma unroll
        for (int i = 0; i < 8; ++i)
            lrow[i] = lrow[i] * fac[i] + rsum[i];
#pragma unroll
        for (int ct = 0; ct < 8; ++ct) {
#pragma unroll
            for (int i = 0; i < 8; ++i) acc[ct][i] *= fac[i];
        }

        // ---- P: C-layout -> A-layout via per-wave LDS scratch. ----
        // Same-wave LDS ops execute in order; scheduling fence is enough.
#pragma unroll
        for (int i = 0; i < 8; ++i) {
            pp[(i + 8 * lhalf) * KTILE + lmod]      = p0[i];
            pp[(i + 8 * lhalf) * KTILE + 16 + lmod] = p1[i];
        }
        __builtin_amdgcn_wave_barrier();

        v2f pa[PSTEPS];
#pragma unroll
        for (int kk = 0; kk < PSTEPS; ++kk)
            pa[kk] = *(const v2f*)(pp + lmod * KTILE + 4 * kk + 2 * lhalf);

        // ---- O += P . V : 8 column tiles x 8 K-steps (row-major V). ----
#pragma unroll
        for (int ct = 0; ct < 8; ++ct) {
            const float* vb = &sV[cur][2 * lhalf * Dn + ct * 16 + lmod];
#pragma unroll
            for (int kk = 0; kk < PSTEPS; ++kk) {
                v2f bf;
                bf.x = vb[(4 * kk + 0) * Dn];
                bf.y = vb[(4 * kk + 1) * Dn];
                acc[ct] = __builtin_amdgcn_wmma_f32_16x16x4_f32(
                              false, pa[kk], false, bf, (short)0, acc[ct],
                              false, false);
            }
        }

        // ---- Handoff: DMA must land before anyone reads the next buffer. --
        if (wave == 0 && have_next)
            __builtin_amdgcn_s_wait_tensorcnt(0);
        __syncthreads();
    }

    // ---- Epilogue: divide by softmax denominator and store. ----
    float inv[8];
#pragma unroll
    for (int i = 0; i < 8; ++i) inv[i] = 1.0f / lrow[i];

    float* obase = Op + ((size_t)b * Qn + q0) * Dn;
#pragma unroll
    for (int ct = 0; ct < 8; ++ct) {
#pragma unroll
        for (int i = 0; i < 8; ++i) {
            const int row = i + 8 * lhalf;
            obase[(size_t)row * Dn + ct * 16 + lmod] = acc[ct][i] * inv[i];
        }
    }
}

extern "C" void kernel_launch(void* const* d_in, const int* in_sizes, int n_in,
                              void* d_out, int out_size, void* d_ws, size_t ws_size,
                              hipStream_t stream) {
    const float* q  = (const float*)d_in[0];
    const float* k  = (const float*)d_in[1];
    const float* v  = (const float*)d_in[2];
    const int*   vl = (const int*)d_in[3];
    float* out = (float*)d_out;

    dim3 grid(Bn * (Qn / QTILE));   // 8 * 16 = 128 workgroups
    dim3 block(256);                // 8 wave32s
    fattn_f32_wmma<<<grid, block, 0, stream>>>(q, k, v, vl, out);
}